// emitter_lightAccu_18528488915175
// MI455X (gfx1250) — compile-verified
//
#include <hip/hip_runtime.h>
#include <cstdint>
#include <cstddef>

#define B_   2
#define IMH  240
#define IMW  320
#define EROW 120
#define ECOL 160
#define EH   8
#define EW   16
#define NG   384                       // 6*8*8 grid centers
#define CHSTRIDE ((size_t)NG*EROW*ECOL)
#define ACCN   (B_*3*NG*EROW*ECOL)     // 44,236,800
#define MASKN  (B_*EROW*ECOL)          // 38,400
#define PI_F 3.14159265358979323846f

// ---- CDNA5 async global->LDS staging (guarded; fallback = sync copy) ----
#if defined(__HIP_DEVICE_COMPILE__) && __has_builtin(__builtin_amdgcn_global_load_async_to_lds_b128)
#define USE_ASYNC 1
#else
#define USE_ASYNC 0
#endif

typedef int v4i_cp __attribute__((vector_size(4 * sizeof(int))));
typedef __attribute__((address_space(1))) v4i_cp* g_v4i_ptr;
typedef __attribute__((address_space(3))) v4i_cp* l_v4i_ptr;

__device__ __forceinline__ void stage16(const float* g, float* l) {
#if USE_ASYNC
  __builtin_amdgcn_global_load_async_to_lds_b128(
      (g_v4i_ptr)(void*)g, (l_v4i_ptr)(void*)l, 0, 0);
#else
  float4 v = *(const float4*)g;
  *(float4*)l = v;
#endif
}

__device__ __forceinline__ void stage_wait() {
#if USE_ASYNC
#if __has_builtin(__builtin_amdgcn_s_wait_asynccnt)
  __builtin_amdgcn_s_wait_asynccnt(0);
#else
  asm volatile("s_wait_asynccnt 0" ::: "memory");
#endif
#endif
}

__device__ __forceinline__ float wave_sum32(float v) {
#pragma unroll
  for (int off = 16; off; off >>= 1) v += __shfl_xor(v, off, 32);
  return v;
}

// ---- zero the reduction workspace (sums + mask counts) ----
__global__ void k_init(float* __restrict__ zbase) {
  int t = blockIdx.x * blockDim.x + threadIdx.x;
  if (t < B_ * 3 * NG + B_) zbase[t] = 0.0f;
}

// ---- grid centers: ctr'' = M12^T (R^T ctr), one thread per (b,n) ----
__global__ void k_centers(const float* __restrict__ layout,
                          const float* __restrict__ camR,
                          float* __restrict__ ctr) {
  int t = blockIdx.x * blockDim.x + threadIdx.x;
  if (t >= B_ * NG) return;
  int b = t / NG, n = t % NG;
  int f = n >> 6, ii = (n >> 3) & 7, jj = n & 7;
  const int t0[6] = {3, 7, 4, 6, 7, 7};
  const int t1[6] = {2, 6, 5, 2, 6, 3};
  const int t2[6] = {0, 4, 0, 5, 3, 4};
  const float* L = layout + b * 24;
  float ox = L[t0[f]*3+0], oy = L[t0[f]*3+1], oz = L[t0[f]*3+2];
  float b1x = (L[t1[f]*3+0]-ox)*0.125f, b1y = (L[t1[f]*3+1]-oy)*0.125f, b1z = (L[t1[f]*3+2]-oz)*0.125f;
  float b2x = (L[t2[f]*3+0]-ox)*0.125f, b2y = (L[t2[f]*3+1]-oy)*0.125f, b2z = (L[t2[f]*3+2]-oz)*0.125f;
  float wi = (float)ii + 0.5f, wj = (float)jj + 0.5f;
  float cx = b1x*wi + b2x*wj + ox;
  float cy = b1y*wi + b2y*wj + oy;
  float cz = b1z*wi + b2z*wj + oz;
  const float* R = camR + b * 9;
  float px = R[0]*cx + R[3]*cy + R[6]*cz;   // (R^T c)_x
  float py = R[1]*cx + R[4]*cy + R[7]*cz;
  float pz = R[2]*cx + R[5]*cy + R[8]*cz;
  float* o = ctr + (size_t)t * 3;           // apply M12: (z, y, -x)
  o[0] = pz; o[1] = py; o[2] = -px;
}

// ---- main kernel: block = (b, env row, 32-col chunk); 256 thr = 32 px x 8 n-phases ----
__global__ __launch_bounds__(256) void k_main(
    const float* __restrict__ normal, const float* __restrict__ depth,
    const float* __restrict__ env,    const float* __restrict__ camK,
    const float* __restrict__ ctr,    float* __restrict__ sums,
    float* __restrict__ mcnt,         float* __restrict__ out) {
  __shared__ float sEnv[32 * 384];        // 32 px x 3 ch x 128 texels (48 KB)
  __shared__ float sCtr[NG * 3];
  __shared__ float sAinv[32][9];
  __shared__ float sPts[32][4];
  __shared__ float sMask[32];

  const int tid = threadIdx.x;
  const int bid = blockIdx.x;
  const int b   = bid / (EROW * 5);
  const int rem = bid % (EROW * 5);
  const int er  = rem / 5;
  const int c0  = (rem % 5) * 32;

  // Issue async env staging first (overlaps with per-pixel setup below).
#pragma unroll
  for (int k = 0; k < 12; ++k) {
    int q = tid + 256 * k;                 // 3072 x 16B chunks
    int fidx = q << 2;                     // linear float idx: p*384 + ch*128 + e
    int p  = fidx / 384;
    int r2 = fidx % 384;
    int ch = r2 >> 7;
    int e  = r2 & 127;
    const float* g = env + (((((size_t)b*3 + ch)*EROW + er)*ECOL + (c0 + p)) * 128 + e);
    stage16(g, &sEnv[fidx]);
  }
  // Stage centers for this batch.
  for (int idx = tid; idx < NG * 3; idx += 256) sCtr[idx] = ctr[(size_t)b * NG * 3 + idx];

  // Per-pixel setup: camera frame, 3x3 inverse, point, mask (first wave only).
  if (tid < 32) {
    const int p = tid, ec = c0 + p;
    const int v = 2 * er, u = 2 * ec;
    const float* nb = normal + ((size_t)b * 3 * IMH + v) * IMW + u;
    float nx = 0.25f * (nb[0] + nb[1] + nb[IMW] + nb[IMW + 1]);
    const float* n1 = nb + (size_t)IMH * IMW;
    float ny = 0.25f * (n1[0] + n1[1] + n1[IMW] + n1[IMW + 1]);
    const float* n2 = nb + 2 * (size_t)IMH * IMW;
    float nz = 0.25f * (n2[0] + n2[1] + n2[IMW] + n2[IMW + 1]);
    float d2 = fminf(fmaxf(nx*nx + ny*ny + nz*nz, 1e-6f), 1.0f);
    float invd = 1.0f / sqrtf(d2);
    nx *= invd; ny *= invd; nz *= invd;
    // camy = normalize(up - (up.n) n), up = (0,1,0)
    float t = ny;
    float yx = -t * nx, yy = 1.0f - t * ny, yz = -t * nz;
    float yn = fmaxf(sqrtf(yx*yx + yy*yy + yz*yz), 1e-12f);
    yx /= yn; yy /= yn; yz /= yn;
    // camx = -normalize(cross(camy, n))
    float xx = yy * nz - yz * ny;
    float xy = yz * nx - yx * nz;
    float xz = yx * ny - yy * nx;
    float xn = fmaxf(sqrtf(xx*xx + xy*xy + xz*xz), 1e-12f);
    xx = -xx / xn; xy = -xy / xn; xz = -xz / xn;
    // A = [camx camy n] + 1e-6 I ; inverse via cofactors (matches reference)
    float a  = xx + 1e-6f, bb = yx,          c = nx;
    float dd = xy,         e  = yy + 1e-6f,  f = ny;
    float g  = xz,         h  = yz,          i = nz + 1e-6f;
    float c00 = e*i - f*h,  c01 = f*g - dd*i, c02 = dd*h - e*g;
    float c10 = c*h - bb*i, c11 = a*i - c*g,  c12 = bb*g - a*h;
    float c20 = bb*f - c*e, c21 = c*dd - a*f, c22 = a*e - bb*dd;
    float idet = 1.0f / (a*c00 + bb*c01 + c*c02);
    sAinv[p][0] = c00*idet; sAinv[p][1] = c10*idet; sAinv[p][2] = c20*idet;
    sAinv[p][3] = c01*idet; sAinv[p][4] = c11*idet; sAinv[p][5] = c21*idet;
    sAinv[p][6] = c02*idet; sAinv[p][7] = c12*idet; sAinv[p][8] = c22*idet;
    // point from depth
    float K00 = camK[b*9 + 0], K02 = camK[b*9 + 2];
    float f_pix = K00 * (IMW * 0.5f) / K02;
    float z  = -depth[((size_t)b * IMH + v) * IMW + u];
    sPts[p][0] = -((float)u - IMW * 0.5f) / f_pix * z;
    sPts[p][1] =  ((float)v - IMH * 0.5f) / f_pix * z;
    sPts[p][2] = z;
    float mask = (z < -0.1f) ? 1.0f : 0.0f;
    sMask[p] = mask;
    __builtin_nontemporal_store(mask, &out[ACCN + ((size_t)b * EROW + er) * ECOL + ec]);
    float ms = wave_sum32(mask);
    if (p == 0) atomicAdd(&mcnt[b], ms);
  }

  stage_wait();          // s_wait_asynccnt 0 (ASYNCcnt -> 0)
  __syncthreads();

  const int p  = tid & 31;
  const int wv = tid >> 5;
  const float A0 = sAinv[p][0], A1 = sAinv[p][1], A2 = sAinv[p][2];
  const float A3 = sAinv[p][3], A4 = sAinv[p][4], A5 = sAinv[p][5];
  const float A6 = sAinv[p][6], A7 = sAinv[p][7], A8 = sAinv[p][8];
  const float px = sPts[p][0], py = sPts[p][1], pz = sPts[p][2];
  const float m01 = sMask[p] * 0.1f;
  const float* ep = &sEnv[p * 384];

  for (int k = 0; k < 48; ++k) {
    const int n = wv + (k << 3);
    float vx = sCtr[n*3 + 0] - px;
    float vy = sCtr[n*3 + 1] - py;
    float vz = sCtr[n*3 + 2] - pz;
    float lx = A0*vx + A1*vy + A2*vz;
    float ly = A3*vx + A4*vy + A5*vz;
    float lz = A6*vx + A7*vy + A8*vz;
    float invn = 1.0f / sqrtf(lx*lx + ly*ly + lz*lz);
    lx *= invn; ly *= invn; lz *= invn;
    float ct    = fminf(fmaxf(lz, -1.0f), 1.0f);
    float theta = acosf(ct);
    float phi   = atan2f(ly, lx);                    // == atan2(ly/sin, lx/sin)
    float az = phi * (8.0f / PI_F) + 7.5f;           // (phi/2pi + 0.5)*16 - 0.5
    float el = theta * (16.0f / PI_F) - 0.5f;        // theta/pi * 16 - 0.5
    float x = fminf(fmaxf(az, 0.0f), (float)(EW - 1));
    float y = fminf(fmaxf(el, 0.0f), (float)(EH - 1));
    float x0f = floorf(x), y0f = floorf(y);
    float wx = x - x0f, wy = y - y0f;
    int x0 = (int)x0f, y0 = (int)y0f;
    int x1 = x0 + 1 < EW - 1 ? x0 + 1 : EW - 1;
    int y1 = y0 + 1 < EH - 1 ? y0 + 1 : EH - 1;
    float w00 = (1.0f - wy) * (1.0f - wx), w01 = (1.0f - wy) * wx;
    float w10 = wy * (1.0f - wx),          w11 = wy * wx;
    int i00 = (y0 << 4) + x0, i01 = (y0 << 4) + x1;
    int i10 = (y1 << 4) + x0, i11 = (y1 << 4) + x1;
    size_t o = ((((size_t)b * 3) * NG + n) * EROW + er) * ECOL + c0 + p;
#pragma unroll
    for (int chn = 0; chn < 3; ++chn) {
      const float* e = ep + (chn << 7);
      float s = e[i00]*w00 + e[i01]*w01 + e[i10]*w10 + e[i11]*w11;
      float val = s * m01;
      __builtin_nontemporal_store(val, &out[o + (size_t)chn * CHSTRIDE]); // streaming 177MB
      float r = wave_sum32(val);
      if (p == 0) atomicAdd(&sums[(b * 3 + chn) * NG + n], r);
    }
  }
}

// ---- mean = sums / (maskcount + 1e-6), layout (B, 384, 3) ----
__global__ void k_mean(const float* __restrict__ sums, const float* __restrict__ mcnt,
                       float* __restrict__ out) {
  int t = blockIdx.x * blockDim.x + threadIdx.x;
  if (t >= B_ * NG * 3) return;
  int b = t / (NG * 3);
  int rem = t % (NG * 3);
  int n = rem / 3, ch = rem % 3;
  out[(size_t)ACCN + MASKN + t] = sums[(b * 3 + ch) * NG + n] / (mcnt[b] + 1e-6f);
}

extern "C" void kernel_launch(void* const* d_in, const int* in_sizes, int n_in,
                              void* d_out, int out_size, void* d_ws, size_t ws_size,
                              hipStream_t stream) {
  const float* normal = (const float*)d_in[0];   // (2,3,240,320)
  const float* depth  = (const float*)d_in[1];   // (2,240,320)
  const float* env    = (const float*)d_in[2];   // (2,3,120,160,8,16)
  const float* camK   = (const float*)d_in[3];   // (2,3,3)
  const float* camR   = (const float*)d_in[4];   // (2,3,3)
  const float* layout = (const float*)d_in[5];   // (2,8,3)
  float* out = (float*)d_out;
  float* ws  = (float*)d_ws;
  float* centers = ws;                 // B*NG*3   = 2304 floats
  float* sums    = ws + 2304;          // B*3*NG   = 2304 floats
  float* mcnt    = ws + 4608;          // B_       = 2 floats

  k_init   <<<10, 256, 0, stream>>>(sums);                 // zero sums + mcnt (contiguous)
  k_centers<<<3,  256, 0, stream>>>(layout, camR, centers);
  k_main   <<<B_ * EROW * 5, 256, 0, stream>>>(normal, depth, env, camK,
                                               centers, sums, mcnt, out);
  k_mean   <<<9,  256, 0, stream>>>(sums, mcnt, out);
}